// GraphConvolutionModified_79955111182918
// MI455X (gfx1250) — compile-verified
//
#include <hip/hip_runtime.h>

#define FDIM 128

typedef __attribute__((ext_vector_type(2))) float v2f;
typedef __attribute__((ext_vector_type(8))) float v8f;

// ---------------- zero-init workspace (agg + diag) ----------------
__global__ void zero_f32(float* __restrict__ p, int n) {
  int i = blockIdx.x * blockDim.x + threadIdx.x;
  if (i < n) p[i] = 0.0f;
}

// ---------------- edge scatter: wave-per-edge -----------------
// agg[src, :] += w * x[dst, :]; diag[src] += w when src==dst.
// Lane l handles features [4l, 4l+4): coalesced float4 gather of x,
// 4 global f32 atomics into agg (resolved in L2; x and agg both fit in 192MB L2).
__global__ void __launch_bounds__(256) scatter_edges(
    const float* __restrict__ x, const int* __restrict__ esrc,
    const int* __restrict__ edst, const float* __restrict__ ew,
    float* __restrict__ agg, float* __restrict__ diag, int E) {
  int gid = blockIdx.x * blockDim.x + threadIdx.x;
  int e = gid >> 5;
  if (e >= E) return;
  int lane = threadIdx.x & 31;
  int s = esrc[e];
  int d = edst[e];
  float w = ew[e];
  const float4 xv = ((const float4*)(x + (size_t)d * FDIM))[lane];
  float* ap = agg + (size_t)s * FDIM + lane * 4;
  atomicAdd(ap + 0, w * xv.x);
  atomicAdd(ap + 1, w * xv.y);
  atomicAdd(ap + 2, w * xv.z);
  atomicAdd(ap + 3, w * xv.w);
  if (s == d && lane == 0) atomicAdd(diag + s, w);
}

__device__ __forceinline__ float label_mask(int f) {
  // LABEL_IDXS = {0, 5, 17, 42}
  return (f == 0 || f == 5 || f == 17 || f == 42) ? 1.0f : 0.0f;
}

// ---------------- fused h + GEMM + bias + ReLU via WMMA f32 16x16x4 ----------------
// One wave computes one 16-row M-tile across all 128 output columns
// (8 accumulator tiles). h = agg - diag[:,None] * (x * mask) computed in-register.
__global__ void __launch_bounds__(256) fused_gemm(
    const float* __restrict__ agg, const float* __restrict__ diag,
    const float* __restrict__ x, const float* __restrict__ Kw,
    const float* __restrict__ bias, float* __restrict__ out, int mtiles) {
  const int lane = threadIdx.x & 31;
  const int wave = threadIdx.x >> 5;
  const int mt = blockIdx.x * 8 + wave;   // wave-uniform; whole wave exits together
  if (mt >= mtiles) return;

  const int half  = lane >> 4;            // 0: K rows {0,1}; 1: K rows {2,3}
  const int l15   = lane & 15;            // A row within tile / B column
  const int khalf = half * 2;
  const int arow  = mt * 16 + l15;

  const float dg = diag[arow];
  const float* aggp = agg + (size_t)arow * FDIM;
  const float* xp   = x   + (size_t)arow * FDIM;

  v8f acc[8];
#pragma unroll
  for (int u = 0; u < 8; ++u) acc[u] = (v8f){0.f, 0.f, 0.f, 0.f, 0.f, 0.f, 0.f, 0.f};

  for (int k0 = 0; k0 < FDIM; k0 += 4) {
    const int f0 = k0 + khalf;
    // A tile: h[arow, f0], h[arow, f0+1]  (f0 even -> aligned float2 load)
    const float2 av = *(const float2*)(aggp + f0);
    const float2 xv = *(const float2*)(xp + f0);
    v2f a;
    a.x = av.x - dg * label_mask(f0)     * xv.x;
    a.y = av.y - dg * label_mask(f0 + 1) * xv.y;

    const float* kp = Kw + (size_t)f0 * FDIM + l15;  // B rows f0, f0+1 at col l15
#pragma unroll
    for (int u = 0; u < 8; ++u) {
      v2f b;
      b.x = kp[u * 16];
      b.y = kp[u * 16 + FDIM];
      acc[u] = __builtin_amdgcn_wmma_f32_16x16x4_f32(
          /*neg_a=*/false, a, /*neg_b=*/false, b,
          /*c_mod=*/(short)0, acc[u], /*reuse_a=*/false, /*reuse_b=*/false);
    }
  }

  // C/D layout: VGPR r -> M = r + 8*half, N = l15
  const int rbase = mt * 16 + half * 8;
#pragma unroll
  for (int u = 0; u < 8; ++u) {
    const float bv = bias[u * 16 + l15];
#pragma unroll
    for (int r = 0; r < 8; ++r) {
      float v = acc[u][r] + bv;
      out[(size_t)(rbase + r) * FDIM + u * 16 + l15] = v > 0.0f ? v : 0.0f;
    }
  }
}

extern "C" void kernel_launch(void* const* d_in, const int* in_sizes, int n_in,
                              void* d_out, int out_size, void* d_ws, size_t ws_size,
                              hipStream_t stream) {
  const float* x    = (const float*)d_in[0];  // [1, N, 128]
  const int*   esrc = (const int*)  d_in[1];  // [E]
  const int*   edst = (const int*)  d_in[2];  // [E]
  const float* ew   = (const float*)d_in[3];  // [E]
  const float* Kw   = (const float*)d_in[4];  // [128, 128]
  const float* bias = (const float*)d_in[5];  // [128]
  float* out = (float*)d_out;

  const int E = in_sizes[1];
  const int N = in_sizes[0] / FDIM;
  const int mtiles = N / 16;                  // N = 50000 -> 3125

  float* agg  = (float*)d_ws;                 // N*128 floats
  float* diag = agg + (size_t)N * FDIM;       // N floats

  // 1) zero agg + diag
  const int zn = N * FDIM + N;
  zero_f32<<<(zn + 255) / 256, 256, 0, stream>>>(agg, zn);

  // 2) edge scatter (wave per edge)
  const long long sthreads = (long long)E * 32;
  const int sblocks = (int)((sthreads + 255) / 256);
  scatter_edges<<<sblocks, 256, 0, stream>>>(x, esrc, edst, ew, agg, diag, E);

  // 3) fused h + GEMM + ReLU (8 waves/block, wave per 16-row tile)
  fused_gemm<<<(mtiles + 7) / 8, 256, 0, stream>>>(agg, diag, x, Kw, bias, out, mtiles);
}